// GCN_13606456393829
// MI455X (gfx1250) — compile-verified
//
#include <hip/hip_runtime.h>

// ---------------------------------------------------------------------------
// GCN forward on gfx1250 (MI455X):
//   norms -> [scatter-add SpMM -> WMMA GEMM(+bias,+relu)] x3
// Dense GEMMs use V_WMMA_F32_16X16X4_F32 (full f32 precision matrix cores).
// Sparse aggregation uses one wave32 per edge with hardware f32 atomics (L2).
// ---------------------------------------------------------------------------

typedef float v2f __attribute__((ext_vector_type(2)));
typedef float v8f __attribute__((ext_vector_type(8)));

#define DK    128            // inner dim of every GEMM / feature width
#define COLS  32             // output columns per workgroup (2 WMMA tiles/wave)
#define NT    (COLS / 16)
#define SX    (DK + 4)       // LDS row stride for X   (== 4 mod 64 -> no bank conflicts)
#define SWT   (DK + 4)       // LDS col stride for W^T (== 4 mod 64)

// ---------------------------------------------------------------------------
__global__ __launch_bounds__(256) void gcn_zero(float4* __restrict__ p, long n4) {
    long i = (long)blockIdx.x * blockDim.x + threadIdx.x;
    long stride = (long)gridDim.x * blockDim.x;
    for (; i < n4; i += stride) p[i] = make_float4(0.f, 0.f, 0.f, 0.f);
}

__global__ __launch_bounds__(256) void gcn_degree(const int* __restrict__ src,
                                                  const int* __restrict__ dst,
                                                  float* __restrict__ outdeg,
                                                  float* __restrict__ indeg, int E) {
    int e = blockIdx.x * 256 + threadIdx.x;
    if (e < E) {
        unsafeAtomicAdd(&outdeg[src[e]], 1.0f);
        unsafeAtomicAdd(&indeg[dst[e]], 1.0f);
    }
}

__global__ __launch_bounds__(256) void gcn_norm(float* __restrict__ ns,
                                                float* __restrict__ nd, int N) {
    int i = blockIdx.x * 256 + threadIdx.x;
    if (i < N) {
        ns[i] = rsqrtf(fmaxf(ns[i], 1.0f));
        nd[i] = rsqrtf(fmaxf(nd[i], 1.0f));
    }
}

// One wave (32 lanes) per edge: coalesced 512B row gather, 128 f32 atomics to L2.
__global__ __launch_bounds__(256) void gcn_scatter(const float* __restrict__ h,
                                                   const float* __restrict__ norm_src,
                                                   const int* __restrict__ src,
                                                   const int* __restrict__ dst,
                                                   float* __restrict__ agg, int E) {
    long tid = (long)blockIdx.x * 256 + threadIdx.x;
    long e   = tid >> 5;
    int lane = threadIdx.x & 31;
    if (e >= E) return;
    int s = src[e];
    int d = dst[e];
    float sc = norm_src[s];
    float4 v = *(const float4*)(h + (long)s * DK + lane * 4);
    float* base = agg + (long)d * DK + lane * 4;
    unsafeAtomicAdd(base + 0, v.x * sc);
    unsafeAtomicAdd(base + 1, v.y * sc);
    unsafeAtomicAdd(base + 2, v.z * sc);
    unsafeAtomicAdd(base + 3, v.w * sc);
}

// out[N,DO] = relu?( (X * norm_dst[:,None]) @ W + b )
// Grid: (ceil(N/64), DO/COLS). Block: 128 threads = 4 waves; wave w owns rows
// [blk*64 + 16w, +16) and COLS output columns. f32 WMMA, K stepped by 4.
__global__ __launch_bounds__(128) void gcn_gemm(const float* __restrict__ X,
                                                const float* __restrict__ norm_dst,
                                                const float* __restrict__ W,
                                                const float* __restrict__ bias,
                                                float* __restrict__ out,
                                                float* __restrict__ out2,
                                                int N, int DO, int relu) {
    __shared__ float ldsX[64 * SX];      // 33792 B : 64-row scaled X tile
    __shared__ float ldsWt[COLS * SWT];  // 16896 B : W panel, transposed [col][k]

    const int tid  = threadIdx.x;
    const int wave = tid >> 5;
    const int lane = tid & 31;
    const long rowBlock = (long)blockIdx.x * 64;
    const int colBase   = blockIdx.y * COLS;

    // Stage W panel transposed: ldsWt[c*SWT + k] = W[k][colBase+c]
    for (int i = tid; i < DK * COLS; i += 128) {
        int k = i >> 5;       // COLS == 32
        int c = i & 31;
        ldsWt[c * SWT + k] = W[(long)k * DO + colBase + c];
    }
    // Stage X tile scaled by norm_dst (coalesced float4 loads)
    for (int i = tid; i < 64 * (DK / 4); i += 128) {
        int r  = i >> 5;      // DK/4 == 32
        int c4 = i & 31;
        long row = rowBlock + r;
        float4 v = make_float4(0.f, 0.f, 0.f, 0.f);
        float s = 0.f;
        if (row < N) {
            v = *(const float4*)(X + row * (long)DK + c4 * 4);
            s = norm_dst[row];
        }
        float4 sv = make_float4(v.x * s, v.y * s, v.z * s, v.w * s);
        *(float4*)(&ldsX[r * SX + c4 * 4]) = sv;
    }
    __syncthreads();

    const long row0 = rowBlock + wave * 16;
    if (row0 >= N) return;   // uniform per wave; EXEC stays all-1s for WMMA

    const int m  = lane & 15;          // A: row within tile / B: col within tile
    const int kb = (lane >> 4) << 1;   // K sub-offset 0 or 2 per ISA layout

    v8f acc[NT];
#pragma unroll
    for (int t = 0; t < NT; ++t) acc[t] = (v8f){0.f, 0.f, 0.f, 0.f, 0.f, 0.f, 0.f, 0.f};

    const float* xrow = &ldsX[(wave * 16 + m) * SX];
    const float* wcol = &ldsWt[m * SWT];

#pragma unroll
    for (int k = 0; k < DK; k += 4) {
        v2f a = *(const v2f*)(xrow + k + kb);   // A: X[row0+m][k+kb .. +1]
#pragma unroll
        for (int t = 0; t < NT; ++t) {
            v2f b = *(const v2f*)(wcol + t * 16 * SWT + k + kb);  // B: W[k+kb..+1][col]
            acc[t] = __builtin_amdgcn_wmma_f32_16x16x4_f32(
                false, a, false, b, (short)0, acc[t], false, false);
        }
    }

    // Epilogue: C/D layout -> VGPR v, half = lane>>4 : M = v + 8*half, N = m
    const int half = lane >> 4;
#pragma unroll
    for (int t = 0; t < NT; ++t) {
        int col = colBase + t * 16 + m;
        float bv = bias[col];
#pragma unroll
        for (int v = 0; v < 8; ++v) {
            long row = row0 + v + 8 * half;
            float val = acc[t][v] + bv;
            if (relu) val = fmaxf(val, 0.f);
            out[row * (long)DO + col] = val;
            if (out2) out2[row * (long)DO + col] = val;
        }
    }
}

// ---------------------------------------------------------------------------
extern "C" void kernel_launch(void* const* d_in, const int* in_sizes, int n_in,
                              void* d_out, int out_size, void* d_ws, size_t ws_size,
                              hipStream_t stream) {
    (void)n_in; (void)out_size; (void)ws_size;
    const float* feats = (const float*)d_in[0];
    const float* W0    = (const float*)d_in[1];
    const float* b0    = (const float*)d_in[2];
    const float* W1    = (const float*)d_in[3];
    const float* b1    = (const float*)d_in[4];
    const float* W2    = (const float*)d_in[5];
    const float* b2    = (const float*)d_in[6];
    const int*   src   = (const int*)d_in[7];
    const int*   dst   = (const int*)d_in[8];

    const int N  = in_sizes[0] / DK;   // 100000
    const int E  = in_sizes[7];        // 1600000
    const int DO2 = in_sizes[6];       // 64 (b2 length)

    float* ws       = (float*)d_ws;
    float* norm_src = ws;                       // N floats (deg, then rsqrt)
    float* norm_dst = ws + N;                   // N floats
    float* agg      = ws + 2 * (long)N;         // N*DK floats

    float* h0  = (float*)d_out;
    float* h1  = h0 + (long)N * DK;
    float* h2  = h1 + (long)N * DK;
    float* h2b = h2 + (long)N * DO2;

    // Zero norms + agg in one pass (N*(DK+2) floats, divisible by 4)
    gcn_zero<<<2048, 256, 0, stream>>>((float4*)ws, (long)N * (DK + 2) / 4);

    // Degree norms
    gcn_degree<<<(E + 255) / 256, 256, 0, stream>>>(src, dst, norm_src, norm_dst, E);
    gcn_norm<<<(N + 255) / 256, 256, 0, stream>>>(norm_src, norm_dst, N);

    const int sblocks = (int)(((long)E * 32 + 255) / 256);
    dim3 ggrid((N + 63) / 64, DK / COLS);

    // Layer 0
    gcn_scatter<<<sblocks, 256, 0, stream>>>(feats, norm_src, src, dst, agg, E);
    gcn_gemm<<<ggrid, 128, 0, stream>>>(agg, norm_dst, W0, b0, h0, nullptr, N, DK, 1);

    // Layer 1
    gcn_zero<<<2048, 256, 0, stream>>>((float4*)agg, (long)N * DK / 4);
    gcn_scatter<<<sblocks, 256, 0, stream>>>(h0, norm_src, src, dst, agg, E);
    gcn_gemm<<<ggrid, 128, 0, stream>>>(agg, norm_dst, W1, b1, h1, nullptr, N, DK, 1);

    // Layer 2 (no relu, duplicate output h2 twice per reference tuple)
    gcn_zero<<<2048, 256, 0, stream>>>((float4*)agg, (long)N * DK / 4);
    gcn_scatter<<<sblocks, 256, 0, stream>>>(h1, norm_src, src, dst, agg, E);
    dim3 g2((N + 63) / 64, DO2 / COLS);
    gcn_gemm<<<g2, 128, 0, stream>>>(agg, norm_dst, W2, b2, h2, h2b, N, DO2, 0);
}